// FocalLoss_22960895165043
// MI455X (gfx1250) — compile-verified
//
#include <hip/hip_runtime.h>
#include <math.h>

#define BB 8
#define MM 64
#define AA 49104
#define CC 80

typedef float v2f __attribute__((ext_vector_type(2)));
typedef float v4f __attribute__((ext_vector_type(4)));
typedef float v8f __attribute__((ext_vector_type(8)));

// ---------------------------------------------------------------------------
// Wave-level fp32 sum using V_WMMA_F32_16X16X4_F32.
// A = ones(16x4), B = [acc0;acc1] packed as 4x16 f32 (K0=acc0 lanes0-15,
// K1=acc0 lanes16-31, K2=acc1 lanes0-15, K3=acc1 lanes16-31).
// D[m,n] = colsum(n); every lane's d[0] holds colsum(lane%16), so a 4-step
// butterfly over 16 lanes finishes the 32-lane reduction. EXEC must be all 1s.
// ---------------------------------------------------------------------------
__device__ __forceinline__ float wave_sum_wmma(float acc0, float acc1) {
    v2f a; a[0] = 1.0f; a[1] = 1.0f;
    v2f bm; bm[0] = acc0; bm[1] = acc1;
    v8f c = {};
    v8f d = __builtin_amdgcn_wmma_f32_16x16x4_f32(
        /*neg_a=*/false, a, /*neg_b=*/false, bm,
        /*c_mod=*/(short)0, c, /*reuse_a=*/false, /*reuse_b=*/false);
    float v = d[0];
    v += __shfl_xor(v, 1, 32);
    v += __shfl_xor(v, 2, 32);
    v += __shfl_xor(v, 4, 32);
    v += __shfl_xor(v, 8, 32);
    return v;  // total in every lane
}

// ---------------------------------------------------------------------------
// Kernel 0: zero the per-image accumulators.
// ---------------------------------------------------------------------------
__global__ void fl_init_kernel(int* __restrict__ npos, float* __restrict__ regsum,
                               float* __restrict__ clssum) {
    int t = threadIdx.x;
    if (t < BB) { npos[t] = 0; regsum[t] = 0.0f; clssum[t] = 0.0f; }
}

// ---------------------------------------------------------------------------
// Kernel 1: anchor->GT assignment, pos mask, npos and smooth-L1 sum.
// One thread per (image, anchor). GT boxes staged in LDS.
// ---------------------------------------------------------------------------
__global__ void fl_assign_kernel(const float* __restrict__ boxes,
                                 const int* __restrict__ labels,
                                 const float* __restrict__ anchors,
                                 const float* __restrict__ regression,
                                 int* __restrict__ tgt,
                                 int* __restrict__ npos,
                                 float* __restrict__ regsum) {
    __shared__ float4 sbox[MM];
    __shared__ float  sarea[MM];
    __shared__ int    slab[MM];
    const int b   = blockIdx.y;
    const int tid = threadIdx.x;
    if (tid < MM) {
        const float* bp = boxes + (size_t)(b * MM + tid) * 4;
        float4 bx = make_float4(bp[0], bp[1], bp[2], bp[3]);  // x1,y1,x2,y2
        sbox[tid]  = bx;
        sarea[tid] = (bx.z - bx.x) * (bx.w - bx.y);
        slab[tid]  = labels[b * MM + tid];
    }
    __syncthreads();

    const int a = blockIdx.x * blockDim.x + tid;
    float localReg = 0.0f;
    int   localPos = 0;
    if (a < AA) {
        const float4 an = *(const float4*)(anchors + (size_t)a * 4);  // y1,x1,y2,x2
        const float ay1 = an.x, ax1 = an.y, ay2 = an.z, ax2 = an.w;
        const float aw = ax2 - ax1, ah = ay2 - ay1;
        const float aarea = aw * ah;
        float best = -1e30f;
        int   bm   = 0;
#pragma unroll 8
        for (int m = 0; m < MM; ++m) {
            float4 bx = sbox[m];
            float iw = fminf(ax2, bx.z) - fmaxf(ax1, bx.x);
            float ih = fminf(ay2, bx.w) - fmaxf(ay1, bx.y);
            iw = fmaxf(iw, 0.0f);
            ih = fmaxf(ih, 0.0f);
            float inter = iw * ih;
            float ua  = fmaxf(aarea + sarea[m] - inter, 1e-8f);
            float iou = inter / ua;
            float v   = (slab[m] != 0) ? iou : -1.0f;
            if (v > best) { best = v; bm = m; }  // strict '>' => first max (JAX argmax)
        }
        const float4 ab = sbox[bm];
        const int    al = slab[bm] - 1;
        const float gw0 = ab.z - ab.x, gh0 = ab.w - ab.y;
        const bool  big = gw0 * gh0 > 100.0f;
        const bool  pos = (big && best >= 0.5f) || (!big && best >= 0.15f);
        tgt[(size_t)b * AA + a] = pos ? al : -1;
        if (pos) {
            localPos = 1;
            const float gcx = ab.x + 0.5f * gw0;
            const float gcy = ab.y + 0.5f * gh0;
            const float gw  = fmaxf(gw0, 1.0f);
            const float gh  = fmaxf(gh0, 1.0f);
            const float acx = ax1 + 0.5f * aw;
            const float acy = ay1 + 0.5f * ah;
            const float t0 = (gcy - acy) / ah;
            const float t1 = (gcx - acx) / aw;
            const float t2 = logf(gh / ah);
            const float t3 = logf(gw / aw);
            const float4 r = *(const float4*)(regression + (size_t)(b * AA + a) * 4);
            const float d0 = fabsf(t0 - r.x), d1 = fabsf(t1 - r.y);
            const float d2 = fabsf(t2 - r.z), d3 = fabsf(t3 - r.w);
            const float th = 1.0f / 9.0f, cth = 0.5f / 9.0f;
            localReg  = (d0 <= th) ? 4.5f * d0 * d0 : d0 - cth;
            localReg += (d1 <= th) ? 4.5f * d1 * d1 : d1 - cth;
            localReg += (d2 <= th) ? 4.5f * d2 * d2 : d2 - cth;
            localReg += (d3 <= th) ? 4.5f * d3 * d3 : d3 - cth;
        }
    }
    // wave32 reduction, one atomic per wave
    for (int off = 16; off > 0; off >>= 1) {
        localReg += __shfl_xor(localReg, off, 32);
        localPos += __shfl_xor(localPos, off, 32);
    }
    if ((tid & 31) == 0) {
        if (localPos)          atomicAdd(&npos[b],   localPos);
        if (localReg != 0.0f)  atomicAdd(&regsum[b], localReg);
    }
}

// ---------------------------------------------------------------------------
// Kernel 2: the 125.7 MB streaming focal pass. b128 non-temporal loads,
// 4 classes / thread / iteration, WMMA fp32 wave reduction at the end.
// ---------------------------------------------------------------------------
__device__ __forceinline__ float focal_term(float x, bool is_one) {
    const float c  = fminf(fmaxf(x, 1e-4f), 1.0f - 1e-4f);
    const float om = 1.0f - c;
    const float pt = is_one ? om : c;             // (is_one ? 1-cls : cls)
    const float w  = (is_one ? 0.25f : 0.75f) * pt * pt;  // alpha * pt^GAMMA
    const float l  = -logf(is_one ? c : om);      // BCE term (targets in {0,1})
    return w * l;
}

__global__ void fl_focal_kernel(const float* __restrict__ cls,
                                const int* __restrict__ tgt,
                                float* __restrict__ clssum) {
    const int b = blockIdx.y;
    const float* cb = cls + (size_t)b * AA * CC;
    const int*   tb = tgt + (size_t)b * AA;
    const int N4 = AA * (CC / 4);          // float4 count per image (CC%4==0)
    const int stride = gridDim.x * blockDim.x;
    float acc0 = 0.0f, acc1 = 0.0f;
    for (int i = blockIdx.x * blockDim.x + threadIdx.x; i < N4; i += stride) {
        v4f v = __builtin_nontemporal_load((const v4f*)cb + i);
        const int a  = i / (CC / 4);               // anchor index
        const int c0 = (i - a * (CC / 4)) * 4;     // first class of this float4
        const int t  = tb[a];                      // -1 or assigned class
        acc0 += focal_term(v[0], (c0 + 0) == t);
        acc0 += focal_term(v[1], (c0 + 1) == t);
        acc1 += focal_term(v[2], (c0 + 2) == t);
        acc1 += focal_term(v[3], (c0 + 3) == t);
    }
    // All lanes reconverged here; EXEC all-1s for the WMMA reduction.
    const float total = wave_sum_wmma(acc0, acc1);
    if ((threadIdx.x & 31) == 0) atomicAdd(&clssum[b], total);
}

// ---------------------------------------------------------------------------
// Kernel 3: finalize means.
// ---------------------------------------------------------------------------
__global__ void fl_finalize_kernel(const int* __restrict__ npos,
                                   const float* __restrict__ regsum,
                                   const float* __restrict__ clssum,
                                   float* __restrict__ out) {
    if (threadIdx.x == 0 && blockIdx.x == 0) {
        float cl = 0.0f, rl = 0.0f;
        for (int j = 0; j < BB; ++j) {
            const float np = (float)npos[j];
            const float d  = fmaxf(np, 1.0f);
            cl += clssum[j] / d;
            rl += (np > 0.0f) ? regsum[j] / (4.0f * d) : 0.0f;
        }
        out[0] = cl / (float)BB;
        out[1] = rl / (float)BB * 50.0f;
    }
}

extern "C" void kernel_launch(void* const* d_in, const int* in_sizes, int n_in,
                              void* d_out, int out_size, void* d_ws, size_t ws_size,
                              hipStream_t stream) {
    const float* boxes          = (const float*)d_in[0];
    const int*   labels         = (const int*)  d_in[1];
    const float* anchors        = (const float*)d_in[2];
    const float* classification = (const float*)d_in[3];
    const float* regression     = (const float*)d_in[4];
    float*       out            = (float*)d_out;

    char* ws = (char*)d_ws;
    size_t tgtBytes = ((size_t)BB * AA * sizeof(int) + 255) & ~(size_t)255;
    int*   tgt    = (int*)ws;
    int*   npos   = (int*)  (ws + tgtBytes);
    float* regsum = (float*)(ws + tgtBytes + 64);
    float* clssum = (float*)(ws + tgtBytes + 128);

    fl_init_kernel<<<1, 32, 0, stream>>>(npos, regsum, clssum);

    dim3 g1((AA + 255) / 256, BB);
    fl_assign_kernel<<<g1, 256, 0, stream>>>(boxes, labels, anchors, regression,
                                             tgt, npos, regsum);

    dim3 g2(1024, BB);  // grid-stride: ~4 float4s/thread of the 982k per image
    fl_focal_kernel<<<g2, 256, 0, stream>>>(classification, tgt, clssum);

    fl_finalize_kernel<<<1, 32, 0, stream>>>(npos, regsum, clssum, out);
}